// GammaLambdaLearner_15178414424613
// MI455X (gfx1250) — compile-verified
//
#include <hip/hip_runtime.h>

#define EPSV 1e-8f

constexpr int T_TOTAL  = 4096;   // input_seq_len + td_extension_steps (reference constants)
constexpr int TAIL     = 2048;   // both concat halves alias lambdas[-2048:]
constexpr int NTHREADS = 256;    // 8 wave32s, single workgroup
constexpr int CHUNK    = 16;     // elements per thread: 256*16 = 4096

typedef __attribute__((ext_vector_type(2))) float v2f;
typedef __attribute__((ext_vector_type(8))) float v8f;

#if defined(__has_builtin)
#if __has_builtin(__builtin_amdgcn_wmma_f32_16x16x4_f32)
#define HAVE_WMMA_F32X4 1
#endif
#endif

__global__ __launch_bounds__(NTHREADS)
void gll_scan_kernel(const float* __restrict__ raw_gamma,
                     const float* __restrict__ raw_lambd,
                     float* __restrict__ out)
{
    __shared__ float sWaveA[NTHREADS / 32];
    __shared__ float sWaveB[NTHREADS / 32];
    __shared__ float sExclA[NTHREADS / 32];
    __shared__ float sExclB[NTHREADS / 32];
    __shared__ float sPart[NTHREADS];
    __shared__ float sScale;

    const int tid  = threadIdx.x;
    const int lane = tid & 31;
    const int wv   = tid >> 5;

    const float gamma = fmaxf(tanhf(raw_gamma[0]), EPSV);

    // ---- pass 1: per-element coefficients + thread-local affine composite ----
    // step s (s ascending = t descending): v -> a*v + (gamma - a), a = gamma*lambda
    float a[CHUNK];
    float Ac = 1.0f, Bc = 0.0f;            // composite of this thread's CHUNK maps
    const int sBase = tid * CHUNK;
    #pragma unroll
    for (int e = 0; e < CHUNK; ++e) {
        const int s   = sBase + e;
        const int t   = (T_TOTAL - 1) - s;
        const int idx = TAIL + (t & (TAIL - 1));
        const float l  = fmaxf(tanhf(raw_lambd[idx]), EPSV);
        const float ae = gamma * l;
        a[e] = ae;
        // apply f_e after current composite: (A,B) -> (ae*A, ae*B + (gamma-ae))
        Bc = fmaf(ae, Bc, gamma - ae);
        Ac = ae * Ac;
    }

    // ---- wave32 inclusive scan of affine composites (Hillis–Steele) ----
    #pragma unroll
    for (int off = 1; off < 32; off <<= 1) {
        const float Ap = __shfl_up(Ac, off, 32);
        const float Bp = __shfl_up(Bc, off, 32);
        if (lane >= off) {                 // prev applied first, then current
            Bc = fmaf(Ac, Bp, Bc);
            Ac = Ac * Ap;
        }
    }
    if (lane == 31) { sWaveA[wv] = Ac; sWaveB[wv] = Bc; }
    __syncthreads();

    // ---- exclusive scan of the 8 wave totals (serial, trivial) ----
    if (tid == 0) {
        float Ea = 1.0f, Eb = 0.0f;
        #pragma unroll
        for (int w = 0; w < NTHREADS / 32; ++w) {
            sExclA[w] = Ea; sExclB[w] = Eb;
            const float Fa = sWaveA[w], Fb = sWaveB[w];
            Eb = fmaf(Fa, Eb, Fb);         // E first, then F
            Ea = Fa * Ea;
        }
    }
    __syncthreads();

    // thread-exclusive composite = laneExclusive ∘ waveExclusive ; v_in = P(1)
    float LA = __shfl_up(Ac, 1, 32);
    float LB = __shfl_up(Bc, 1, 32);
    if (lane == 0) { LA = 1.0f; LB = 0.0f; }
    const float WA = sExclA[wv], WB = sExclB[wv];
    const float PB = fmaf(LA, WB, LB);
    const float PA = LA * WA;
    float v = PA + PB;                     // initial carry v0 = 1.0

    // ---- pass 2: replay the CHUNK serial steps, produce w, local sum ----
    float wvls[CHUNK];
    float lsum = 0.0f;
    #pragma unroll
    for (int e = 0; e < CHUNK; ++e) {
        v = fmaf(a[e], v, gamma - a[e]);
        const float we = fmaxf(1.0f - v, EPSV);
        wvls[e] = we;
        lsum += we;
    }
    sPart[tid] = lsum;
    __syncthreads();

    // ---- reduce 256 partials -> mean (matrix-unit reduction on wave 0) ----
    if (wv == 0) {
        float total;
#ifdef HAVE_WMMA_F32X4
        // Treat sPart as four 16x4 f32 A-tiles; B = ones(4x16) => D rows = row-sums.
        // A layout (ISA 7.12.2): lanes 0-15 hold K=0,1; lanes 16-31 hold K=2,3; M = lane%16.
        v8f acc = {};
        const v2f ones = {1.0f, 1.0f};
        const int m  = lane & 15;
        const int kh = (lane >> 4) << 1;
        #pragma unroll
        for (int i = 0; i < 4; ++i) {
            v2f amat;
            amat.x = sPart[i * 64 + m * 4 + kh + 0];
            amat.y = sPart[i * 64 + m * 4 + kh + 1];
            acc = __builtin_amdgcn_wmma_f32_16x16x4_f32(
                false, amat, false, ones, (short)0, acc, false, false);
        }
        // D layout: lane holds M = r + 8*(lane>>4), r=0..7 (any column N; all equal).
        float s8 = acc[0] + acc[1] + acc[2] + acc[3]
                 + acc[4] + acc[5] + acc[6] + acc[7];
        total = s8 + __shfl_xor(s8, 16, 32);   // S0..7 + S8..15
#else
        float s8 = 0.0f;
        #pragma unroll
        for (int i = 0; i < 8; ++i) s8 += sPart[lane * 8 + i];
        #pragma unroll
        for (int off = 16; off >= 1; off >>= 1) s8 += __shfl_xor(s8, off, 32);
        total = s8;
#endif
        if (lane == 0) {
            const float mean = total * (1.0f / (float)T_TOTAL);
            sScale = 1.0f / fmaxf(mean, EPSV);
        }
    }
    __syncthreads();

    // ---- normalize and store: output index is t (reverse of scan order s) ----
    const float scale = sScale;
    #pragma unroll
    for (int e = 0; e < CHUNK; ++e) {
        const int s = sBase + e;
        const int t = (T_TOTAL - 1) - s;
        out[t] = wvls[e] * scale;
    }
}

extern "C" void kernel_launch(void* const* d_in, const int* in_sizes, int n_in,
                              void* d_out, int out_size, void* d_ws, size_t ws_size,
                              hipStream_t stream) {
    (void)in_sizes; (void)n_in; (void)out_size; (void)d_ws; (void)ws_size;
    const float* raw_gamma = (const float*)d_in[0];
    const float* raw_lambd = (const float*)d_in[1];
    float* out = (float*)d_out;
    gll_scan_kernel<<<1, NTHREADS, 0, stream>>>(raw_gamma, raw_lambd, out);
}